// ConsistencyLoss_28613072126086
// MI455X (gfx1250) — compile-verified
//
#include <hip/hip_runtime.h>

#define B_   16
#define N_   2
#define H_   512
#define W_   768
#define HW_  (H_ * W_)
#define TPB  256
#define WPB  8            // waves per block (wave32)
#define EPSF 1e-6f

typedef __attribute__((ext_vector_type(2))) float v2f;
typedef __attribute__((ext_vector_type(8))) float v8f;

// Full wave32 cross-lane sum via one V_WMMA_F32_16X16X4_F32.
// A (16x4): lanes 0-15 hold {K0,K1}, lanes 16-31 hold {K2,K3}. We place the
// per-lane value in component 0 and zero in component 1, B = ones(4x16), C = 0.
// D[i][j] = acc(lane i) + acc(lane 16+i) for every column j. Summing the 8 D
// VGPRs gives (lanes 0-15) sum of rows 0-7 and (lanes 16-31) sum of rows 8-15;
// one xor-16 shuffle completes the 32-lane sum. Requires EXEC == all ones.
__device__ __forceinline__ float wave_sum32(float v) {
    v2f a;   a[0] = v;    a[1] = 0.0f;
    v2f one; one[0] = 1.0f; one[1] = 1.0f;
    v8f c = {};
    v8f d = __builtin_amdgcn_wmma_f32_16x16x4_f32(
        /*neg_a=*/false, a, /*neg_b=*/false, one,
        /*c_mod=*/(short)0, c, /*reuse_a=*/false, /*reuse_b=*/false);
    float s = ((d[0] + d[1]) + (d[2] + d[3])) + ((d[4] + d[5]) + (d[6] + d[7]));
    s += __shfl_xor(s, 16, 32);
    return s;
}

__global__ void consistency_main(const float* __restrict__ depths,
                                 const float* __restrict__ extr,
                                 const float* __restrict__ intr,
                                 const float* __restrict__ flows,
                                 const float* __restrict__ masks,
                                 float* __restrict__ partials,
                                 int wavesPerImg) {
    __shared__ float sMat[9];
    __shared__ float sVec[3];
    __shared__ float sCst[8]; // inv_fx_r, inv_fy_r, cx_r, cy_r, fx_t, fy_t, cx_t, cy_t

    const int bk = blockIdx.z;
    const int b  = bk >> 1;
    const int k  = bk & 1;
    const int j  = 1 - k;

    if (threadIdx.x == 0) {
        const float* Er = extr + (size_t)(b * N_ + k) * 12;
        const float* Et = extr + (size_t)(b * N_ + j) * 12;
        float Rr[3][3], tr[3], Rt[3][3], tt[3];
        for (int r = 0; r < 3; ++r) {
            Rr[r][0] = Er[r*4+0]; Rr[r][1] = Er[r*4+1]; Rr[r][2] = Er[r*4+2]; tr[r] = Er[r*4+3];
            Rt[r][0] = Et[r*4+0]; Rt[r][1] = Et[r*4+1]; Rt[r][2] = Et[r*4+2]; tt[r] = Et[r*4+3];
        }
        // p_cam_t = Rt^T (Rr p + tr - tt)  ->  M = Rt^T Rr ; v = Rt^T (tr - tt)
        for (int i = 0; i < 3; ++i) {
            for (int l = 0; l < 3; ++l)
                sMat[i*3+l] = Rt[0][i]*Rr[0][l] + Rt[1][i]*Rr[1][l] + Rt[2][i]*Rr[2][l];
            sVec[i] = Rt[0][i]*(tr[0]-tt[0]) + Rt[1][i]*(tr[1]-tt[1]) + Rt[2][i]*(tr[2]-tt[2]);
        }
        const float* Ir = intr + (size_t)(b * N_ + k) * 4;
        const float* It = intr + (size_t)(b * N_ + j) * 4;
        sCst[0] = 1.0f / Ir[0]; sCst[1] = 1.0f / Ir[1]; sCst[2] = Ir[2]; sCst[3] = Ir[3];
        sCst[4] = It[0];        sCst[5] = It[1];        sCst[6] = It[2]; sCst[7] = It[3];
    }
    __syncthreads();

    const float m00 = sMat[0], m01 = sMat[1], m02 = sMat[2];
    const float m10 = sMat[3], m11 = sMat[4], m12 = sMat[5];
    const float m20 = sMat[6], m21 = sMat[7], m22 = sMat[8];
    const float v0 = sVec[0], v1 = sVec[1], v2 = sVec[2];
    const float ifx = sCst[0], ify = sCst[1], cxr = sCst[2], cyr = sCst[3];
    const float fxt = sCst[4], fyt = sCst[5], cxt = sCst[6], cyt = sCst[7];

    const float* __restrict__ dref = depths + (size_t)(b * N_ + k) * HW_;
    const float* __restrict__ dtgt = depths + (size_t)(b * N_ + j) * HW_;
    const float* __restrict__ flx  = flows + (size_t)((k * B_ + b) * 2 + 0) * HW_;
    const float* __restrict__ fly  = flows + (size_t)((k * B_ + b) * 2 + 1) * HW_;
    const float* __restrict__ msk  = masks + (size_t)(k * B_ + b) * HW_;

    float aR = 0.f, aD = 0.f, aG = 0.f, aM = 0.f;
    const int stride = gridDim.x * TPB;
    for (int p = blockIdx.x * TPB + threadIdx.x; p < HW_; p += stride) {
        const int y = p / W_;
        const int x = p - y * W_;
        const float d  = dref[p];
        const float fx = flx[p];
        const float fy = fly[p];
        const float m  = msk[p];

        // unproject + fused rigid transform
        const float rx = ((float)x - cxr) * ifx;
        const float ry = ((float)y - cyr) * ify;
        const float px = rx * d, py = ry * d, pz = d;
        const float qx = m00*px + m01*py + m02*pz + v0;
        const float qy = m10*px + m11*py + m12*pz + v1;
        const float qz = m20*px + m21*py + m22*pz + v2;

        // reprojection residual
        const float mx = (float)x + fx;
        const float my = (float)y + fy;
        const float iz = 1.0f / qz;
        const float dxr = qx * iz * fxt + cxt - mx;
        const float dyr = qy * iz * fyt + cyt - my;
        const float rd  = sqrtf(dxr*dxr + dyr*dyr);

        // bilinear gather of target-view depth at matched location
        const float sx = mx - 0.5f, sy = my - 0.5f;
        const float fx0 = floorf(sx), fy0 = floorf(sy);
        const float wx = sx - fx0, wy = sy - fy0;
        int x0 = (int)fx0, y0 = (int)fy0;
        int x1 = x0 + 1,  y1 = y0 + 1;
        x0 = min(max(x0, 0), W_ - 1); x1 = min(max(x1, 0), W_ - 1);
        y0 = min(max(y0, 0), H_ - 1); y1 = min(max(y1, 0), H_ - 1);
        const float d00 = dtgt[y0 * W_ + x0];
        const float d01 = dtgt[y0 * W_ + x1];
        const float d10 = dtgt[y1 * W_ + x0];
        const float d11 = dtgt[y1 * W_ + x1];
        const float dw = (1.f - wy) * ((1.f - wx) * d00 + wx * d01)
                       +        wy  * ((1.f - wx) * d10 + wx * d11);

        // disparity + log depth-ratio terms
        const float disp = fabsf(iz - 1.0f / dw);
        const float adw = fabsf(dw), adt = fabsf(qz);
        const float g = fabsf(logf(fminf(adw, adt) / fmaxf(adw, adt)));

        aR += m * rd;
        aD += m * disp;
        aG += m * g;
        aM += m;
    }

    // EXEC is all-ones here (uniform trip count); WMMA wave reduction
    const float sR = wave_sum32(aR);
    const float sD = wave_sum32(aD);
    const float sG = wave_sum32(aG);
    const float sW = wave_sum32(aM);
    if ((threadIdx.x & 31) == 0) {
        const int wid = blockIdx.x * WPB + (threadIdx.x >> 5);
        float* o = partials + ((size_t)bk * wavesPerImg + wid) * 4;
        o[0] = sR; o[1] = sD; o[2] = sG; o[3] = sW;
    }
}

__global__ void consistency_finalize(const float* __restrict__ partials,
                                     const float* __restrict__ intr,
                                     float* __restrict__ out,
                                     int wavesPerImg) {
    const int t = threadIdx.x;      // 0..31 == bk
    const int b = t >> 1;
    const int k = t & 1;
    float sR = 0.f, sD = 0.f, sG = 0.f, sW = 0.f;
    const float* base = partials + (size_t)t * wavesPerImg * 4;
    for (int w = 0; w < wavesPerImg; ++w) {   // fixed order: deterministic
        sR += base[w*4 + 0];
        sD += base[w*4 + 1];
        sG += base[w*4 + 2];
        sW += base[w*4 + 3];
    }
    const float inv = 1.0f / (sW + EPSF);
    float fk = 0.f;                 // scalar mean of intr[:,k,:2] over batch
    for (int bb = 0; bb < B_; ++bb)
        fk += intr[(size_t)(bb * N_ + k) * 4 + 0] + intr[(size_t)(bb * N_ + k) * 4 + 1];
    fk *= (1.0f / (2.0f * B_));

    const float rp = sR * inv;            // LAMBDA_REPROJ = 1
    const float dp = fk * sD * inv;       // LAMBDA_DISP = 1
    const float gp = sG * inv;            // LAMBDA_DEPTH_RATIO folded into term

    const float rp2 = 0.5f * (rp + __shfl_xor(rp, 1, 32));
    const float dp2 = 0.5f * (dp + __shfl_xor(dp, 1, 32));
    const float gp2 = 0.5f * (gp + __shfl_xor(gp, 1, 32));

    __shared__ float tot[B_];
    if (k == 0) {
        out[1  + b] = rp2;
        out[17 + b] = dp2;
        out[33 + b] = gp2;
        tot[b] = rp2 + dp2 + gp2;
    }
    __syncthreads();
    if (t == 0) {
        float s = 0.f;
        for (int bb = 0; bb < B_; ++bb) s += tot[bb];
        out[0] = s * (1.0f / (float)B_);
    }
}

extern "C" void kernel_launch(void* const* d_in, const int* in_sizes, int n_in,
                              void* d_out, int out_size, void* d_ws, size_t ws_size,
                              hipStream_t stream) {
    const float* depths = (const float*)d_in[0];
    const float* extr   = (const float*)d_in[1];
    const float* intr   = (const float*)d_in[2];
    const float* flows  = (const float*)d_in[3];
    const float* masks  = (const float*)d_in[4];
    float* out      = (float*)d_out;
    float* partials = (float*)d_ws;

    // blocks-per-image: pick the largest divisor of 1536 (= HW/TPB) whose
    // partial buffer fits in ws, so the grid-stride trip count stays uniform
    // (EXEC all-ones at the WMMA reduction).
    int bpi = 1;
    const int cand[7] = {96, 48, 24, 12, 6, 3, 1};
    for (int i = 0; i < 7; ++i) {
        size_t need = (size_t)(B_ * N_) * cand[i] * WPB * 4 * sizeof(float);
        if (need <= ws_size) { bpi = cand[i]; break; }
    }
    const int wavesPerImg = bpi * WPB;

    dim3 grid(bpi, 1, B_ * N_);
    consistency_main<<<grid, TPB, 0, stream>>>(depths, extr, intr, flows, masks,
                                               partials, wavesPerImg);
    consistency_finalize<<<1, 32, 0, stream>>>(partials, intr, out, wavesPerImg);
}